// ImbalanceAwareFocalLoss_14912126452150
// MI455X (gfx1250) — compile-verified
//
#include <hip/hip_runtime.h>
#include <math.h>

#define NUM_CLASSES 1000
#define ROWS_PER_BLOCK 128   // 8 waves * 16 rows

typedef float v2f __attribute__((ext_vector_type(2)));
typedef float v8f __attribute__((ext_vector_type(8)));

// ---------------- Pass 1: histogram of targets (integer atomics: deterministic) -------------
__global__ __launch_bounds__(256) void focal_hist_kernel(const int* __restrict__ targets,
                                                         unsigned* __restrict__ counts, int n) {
    __shared__ unsigned lh[NUM_CLASSES];
    for (int i = threadIdx.x; i < NUM_CLASSES; i += blockDim.x) lh[i] = 0u;
    __syncthreads();
    int idx = blockIdx.x * blockDim.x + threadIdx.x;
    int stride = gridDim.x * blockDim.x;
    for (int i = idx; i < n; i += stride) {
        atomicAdd(&lh[targets[i]], 1u);
    }
    __syncthreads();
    for (int i = threadIdx.x; i < NUM_CLASSES; i += blockDim.x) {
        unsigned v = lh[i];
        if (v) atomicAdd(&counts[i], v);
    }
}

// ---------------- Pass 2: class weights --------------------------------------------------
__global__ void focal_weights_kernel(const unsigned* __restrict__ counts,
                                     float* __restrict__ weights, float nf) {
    int c = blockIdx.x * blockDim.x + threadIdx.x;
    if (c < NUM_CLASSES) {
        float freq = (float)counts[c] / (nf + 1e-6f);
        weights[c] = 1.0f / (log1pf(freq) + 1e-6f);
    }
}

// ---------------- Pass 3: fused log-softmax + focal loss, one wave per row ---------------
// Row (1000 f32 = 4000 B, 16B-aligned) is read ONCE into VGPRs as 8 float4 per lane.
__global__ __launch_bounds__(256) void focal_main_kernel(const float* __restrict__ logits,
                                                         const int* __restrict__ targets,
                                                         const float* __restrict__ weights,
                                                         float* __restrict__ partials, int n) {
    __shared__ float wavesum[8];
    const int lane = threadIdx.x & 31;
    const int wave = threadIdx.x >> 5;
    const float NEG_INF = -__builtin_huge_valf();

    float wsum = 0.0f;
    int row0 = blockIdx.x * ROWS_PER_BLOCK + wave * 16;
    for (int i = 0; i < 16; ++i) {
        int r = row0 + i;
        if (r >= n) break;  // uniform per wave
        size_t base = (size_t)r * NUM_CLASSES;
        const float4* p = (const float4*)(logits + base);

        // gfx1250 prefetch of next row (global_prefetch_b8); 32 lanes * 128B covers 4000B
        if (r + 1 < n) __builtin_prefetch(logits + base + NUM_CLASSES + lane * 32, 0, 1);

        // Cols [0,896): lane handles float4 at col 128*j + 4*lane. Tail [896,1000): 26 float4s.
        float4 v[8];
#pragma unroll
        for (int j = 0; j < 7; ++j) v[j] = p[j * 32 + lane];
        if (lane < 26) {
            v[7] = p[224 + lane];
        } else {
            v[7].x = NEG_INF; v[7].y = NEG_INF; v[7].z = NEG_INF; v[7].w = NEG_INF;
        }

        // Row max (wave32 butterfly)
        float m = NEG_INF;
#pragma unroll
        for (int j = 0; j < 8; ++j)
            m = fmaxf(m, fmaxf(fmaxf(v[j].x, v[j].y), fmaxf(v[j].z, v[j].w)));
#pragma unroll
        for (int off = 16; off > 0; off >>= 1) m = fmaxf(m, __shfl_xor(m, off, 32));

        // Sum of exp (tail lanes contribute exp(-inf)=0)
        float s = 0.0f;
#pragma unroll
        for (int j = 0; j < 8; ++j) {
            s += expf(v[j].x - m); s += expf(v[j].y - m);
            s += expf(v[j].z - m); s += expf(v[j].w - m);
        }
#pragma unroll
        for (int off = 16; off > 0; off >>= 1) s += __shfl_xor(s, off, 32);

        int   t      = targets[r];
        float lt     = logits[base + t];          // uniform address, cache hit
        float logp_t = lt - m - logf(s);
        float pt     = expf(logp_t);
        float om     = 1.0f - pt;
        float focal  = om * om;                   // gamma = 2
        wsum += focal * weights[t] * (-logp_t);   // fixed order -> deterministic
    }
    if (lane == 0) wavesum[wave] = wsum;
    __syncthreads();
    if (threadIdx.x == 0) {
        float bs = 0.0f;
#pragma unroll
        for (int w = 0; w < 8; ++w) bs += wavesum[w];
        partials[blockIdx.x] = bs;
    }
}

// ---------------- Pass 4: deterministic reduction of partials via f32 WMMA ---------------
// D = A x OnesB + C accumulates exact f32 row-sums; grand total = lane0 + lane16 sums.
__global__ void focal_final_kernel(const float* __restrict__ partials, float* __restrict__ out,
                                   int nPadded, float invN) {
    int lane = threadIdx.x;  // exactly one wave32, EXEC all ones
#if defined(__gfx1250__) && __has_builtin(__builtin_amdgcn_wmma_f32_16x16x4_f32)
    v8f c = {};
    v2f b1; b1[0] = 1.0f; b1[1] = 1.0f;
    for (int it = 0; it < nPadded; it += 64) {
        v2f a;
        a[0] = partials[it + 2 * lane];
        a[1] = partials[it + 2 * lane + 1];
        c = __builtin_amdgcn_wmma_f32_16x16x4_f32(false, a, false, b1, (short)0, c,
                                                  false, false);
    }
    float s = c[0] + c[1] + c[2] + c[3] + c[4] + c[5] + c[6] + c[7];
    float total = __shfl(s, 0, 32) + __shfl(s, 16, 32);
#else
    float total = 0.0f;
    if (lane == 0)
        for (int i = 0; i < nPadded; ++i) total += partials[i];
#endif
    if (lane == 0) out[0] = total * invN;
}

// ---------------- Launcher ---------------------------------------------------------------
extern "C" void kernel_launch(void* const* d_in, const int* in_sizes, int n_in,
                              void* d_out, int out_size, void* d_ws, size_t ws_size,
                              hipStream_t stream) {
    const float* logits  = (const float*)d_in[0];
    const int*   targets = (const int*)d_in[1];
    int n = in_sizes[1];

    unsigned* counts   = (unsigned*)d_ws;
    float*    weights  = (float*)((char*)d_ws + 4096);
    float*    partials = (float*)((char*)d_ws + 8192);

    int blocks = (n + ROWS_PER_BLOCK - 1) / ROWS_PER_BLOCK;     // 1024 for n=131072
    int padded = ((blocks + 63) / 64) * 64;

    // zero counts + weights + (padded) partials each call; graph-capture-safe
    hipMemsetAsync(d_ws, 0, 8192 + (size_t)padded * sizeof(float), stream);

    focal_hist_kernel<<<256, 256, 0, stream>>>(targets, counts, n);
    focal_weights_kernel<<<(NUM_CLASSES + 255) / 256, 256, 0, stream>>>(counts, weights, (float)n);
    focal_main_kernel<<<blocks, 256, 0, stream>>>(logits, targets, weights, partials, n);
    focal_final_kernel<<<1, 32, 0, stream>>>(partials, (float*)d_out, padded, 1.0f / (float)n);
}